// KalmanFilter_32469952758082
// MI455X (gfx1250) — compile-verified
//
#include <hip/hip_runtime.h>
#include <math.h>

#define B_ROWS 4096
#define T_LEN  8192
#define QF     1e-5f
#define MED_RANK ((T_LEN - 1) / 2)   // 4095 (lower middle element)

typedef float v2f __attribute__((ext_vector_type(2)));
typedef float v8f __attribute__((ext_vector_type(8)));

#if defined(__has_builtin)
#if __has_builtin(__builtin_amdgcn_wmma_f32_16x16x4_f32)
#define USE_WMMA_F32 1
#endif
#endif

__device__ __forceinline__ unsigned int key_of(float x) {
    unsigned int u = __float_as_uint(x);
    // monotonic transform: order of keys == order of floats
    return (u & 0x80000000u) ? ~u : (u | 0x80000000u);
}

// ---------------------------------------------------------------------------
// Kernel 1: one workgroup per row.
//   - single global read of the row
//   - Sum / SumSq via V_WMMA_F32_16X16X4_F32 ones-reduction (f32 accumulate)
//   - transformed keys staged in LDS, 4x8-bit radix select for the median
// ---------------------------------------------------------------------------
__global__ __launch_bounds__(256)
void stats_median_kernel(const float* __restrict__ in,
                         float* __restrict__ Rout,
                         float* __restrict__ x0out) {
    __shared__ unsigned int s_keys[T_LEN];
    __shared__ unsigned int s_hist[256];
    __shared__ float        s_psum[8];
    __shared__ float        s_psq[8];
    __shared__ unsigned int s_prefix;
    __shared__ unsigned int s_rank;

    const int tid = threadIdx.x;
    const int row = blockIdx.x;
    const float* __restrict__ rp = in + (size_t)row * T_LEN;

#ifdef USE_WMMA_F32
    v8f c0 = {}; v8f c1 = {};      // sum accumulators (pipelined pair)
    v8f q0 = {}; v8f q1 = {};      // sum-of-squares accumulators
    v2f ones; ones.x = 1.0f; ones.y = 1.0f;
#else
    float acc_s = 0.0f, acc_q = 0.0f;
#endif

    #pragma unroll
    for (int i = 0; i < 8; ++i) {
        const int idx = i * 1024 + tid * 4;            // coalesced float4
        float4 v = *reinterpret_cast<const float4*>(rp + idx);
        s_keys[idx + 0] = key_of(v.x);
        s_keys[idx + 1] = key_of(v.y);
        s_keys[idx + 2] = key_of(v.z);
        s_keys[idx + 3] = key_of(v.w);
#ifdef USE_WMMA_F32
        v2f a0; a0.x = v.x;       a0.y = v.y;
        v2f a1; a1.x = v.z;       a1.y = v.w;
        v2f b0; b0.x = v.x * v.x; b0.y = v.y * v.y;
        v2f b1; b1.x = v.z * v.z; b1.y = v.w * v.w;
        // D = A(16x4) x Ones(4x16) + C  ->  row-sums replicated over N
        c0 = __builtin_amdgcn_wmma_f32_16x16x4_f32(false, a0, false, ones, (short)0, c0, false, false);
        c1 = __builtin_amdgcn_wmma_f32_16x16x4_f32(false, a1, false, ones, (short)0, c1, false, false);
        q0 = __builtin_amdgcn_wmma_f32_16x16x4_f32(false, b0, false, ones, (short)0, q0, false, false);
        q1 = __builtin_amdgcn_wmma_f32_16x16x4_f32(false, b1, false, ones, (short)0, q1, false, false);
#else
        acc_s += v.x + v.y + v.z + v.w;
        acc_q += v.x * v.x + v.y * v.y + v.z * v.z + v.w * v.w;
#endif
    }

    const int wid  = tid >> 5;
    const int lane = tid & 31;
#ifdef USE_WMMA_F32
    v8f cs = c0 + c1;
    v8f cq = q0 + q1;
    float s = 0.0f, sq = 0.0f;
    #pragma unroll
    for (int k = 0; k < 8; ++k) { s += cs[k]; sq += cq[k]; }
    // lanes 0..15 hold (M=0..7) half-sum, lanes 16..31 the (M=8..15) half-sum
    float wave_s = __shfl(s, 0, 32) + __shfl(s, 16, 32);
    float wave_q = __shfl(sq, 0, 32) + __shfl(sq, 16, 32);
#else
    float s = acc_s, sq = acc_q;
    #pragma unroll
    for (int o = 16; o > 0; o >>= 1) {
        s  += __shfl_xor(s, o, 32);
        sq += __shfl_xor(sq, o, 32);
    }
    float wave_s = s, wave_q = sq;
#endif
    if (lane == 0) { s_psum[wid] = wave_s; s_psq[wid] = wave_q; }
    __syncthreads();

    if (tid == 0) {
        float S = 0.0f, SQ = 0.0f;
        #pragma unroll
        for (int w = 0; w < 8; ++w) { S += s_psum[w]; SQ += s_psq[w]; }
        float mean = S / (float)T_LEN;
        float var  = (SQ - (float)T_LEN * mean * mean) / (float)(T_LEN - 1);
        Rout[row] = 2.0f * var;                 // R_SCALE = 2.0
        s_prefix = 0u;
        s_rank   = MED_RANK;
    }
    __syncthreads();

    // ---- 4-pass radix select on keys in LDS ----
    for (int shift = 24; shift >= 0; shift -= 8) {
        s_hist[tid] = 0u;
        __syncthreads();
        const unsigned int maskAbove =
            (shift == 24) ? 0u : (0xFFFFFFFFu << (shift + 8));
        const unsigned int pref = s_prefix;
        #pragma unroll
        for (int i = 0; i < 32; ++i) {
            unsigned int k = s_keys[tid + i * 256];
            if ((k & maskAbove) == pref)
                atomicAdd(&s_hist[(k >> shift) & 0xFFu], 1u);
        }
        __syncthreads();
        if (tid == 0) {
            unsigned int r = s_rank, c = 0u, d = 0u;
            for (unsigned int b = 0; b < 256; ++b) {
                unsigned int nc = c + s_hist[b];
                if (r < nc) { d = b; break; }
                c = nc;
            }
            s_prefix = pref | (d << shift);
            s_rank   = r - c;
        }
        __syncthreads();
    }

    if (tid == 0) {
        unsigned int k = s_prefix;
        unsigned int u = (k & 0x80000000u) ? (k ^ 0x80000000u) : ~k;
        x0out[row] = __uint_as_float(u);
    }
}

// ---------------------------------------------------------------------------
// Kernel 2: parallel Kalman scan. One workgroup per row; thread j owns the
// 32 time steps [32j, 32j+32).
//   - chunk-start P via normalized binary exponentiation of the constant
//     Moebius matrix M=[[R,RQ],[1,Q+R]]  (P_t = M^t(P_0), f64, 13 iters)
//   - chunk-local affine composition (A,B), block-wide affine Hillis-Steele
//     scan in LDS, then replay + vectorized stores
// ---------------------------------------------------------------------------
__global__ __launch_bounds__(256)
void kalman_scan_kernel(const float* __restrict__ in,
                        const float* __restrict__ Rin,
                        const float* __restrict__ x0in,
                        float* __restrict__ out) {
    __shared__ float sA[256];
    __shared__ float sB[256];

    const int tid  = threadIdx.x;
    const int row  = blockIdx.x;
    const size_t base = (size_t)row * T_LEN;
    const float Rf = Rin[row];
    const float x0 = x0in[row];
    const int t0 = tid * 32;

    float yv[32];
    #pragma unroll
    for (int i = 0; i < 8; ++i) {
        float4 v = *reinterpret_cast<const float4*>(in + base + t0 + i * 4);
        yv[i * 4 + 0] = v.x; yv[i * 4 + 1] = v.y;
        yv[i * 4 + 2] = v.z; yv[i * 4 + 3] = v.w;
    }

    // ---- P at chunk start: Moebius power M^(t0-1) applied to P0=1 ----
    double P_start = 1.0;
    if (tid != 0) {
        const double Rd = (double)Rf;
        const double Qd = (double)QF;
        double b00 = Rd,  b01 = Rd * Qd;
        double b10 = 1.0, b11 = Qd + Rd;
        double m00 = 1.0, m01 = 0.0, m10 = 0.0, m11 = 1.0;
        unsigned int e = (unsigned int)(t0 - 1);
        while (e) {
            if (e & 1u) {
                double n00 = m00 * b00 + m01 * b10;
                double n01 = m00 * b01 + m01 * b11;
                double n10 = m10 * b00 + m11 * b10;
                double n11 = m10 * b01 + m11 * b11;
                double inv = 1.0 / (n00 + n11);      // scale-invariant Moebius
                m00 = n00 * inv; m01 = n01 * inv;
                m10 = n10 * inv; m11 = n11 * inv;
            }
            e >>= 1;
            if (e) {
                double n00 = b00 * b00 + b01 * b10;
                double n01 = b00 * b01 + b01 * b11;
                double n10 = b10 * b00 + b11 * b10;
                double n11 = b10 * b01 + b11 * b11;
                double inv = 1.0 / (n00 + n11);
                b00 = n00 * inv; b01 = n01 * inv;
                b10 = n10 * inv; b11 = n11 * inv;
            }
        }
        P_start = (m00 + m01) / (m10 + m11);         // P0 = 1
    }

    // ---- pass 1: local affine composition x -> A*x + B over the chunk ----
    float Pf = (float)P_start;
    float Ac = 1.0f, Bc = 0.0f;
    #pragma unroll
    for (int i = 0; i < 32; ++i) {
        const int t = t0 + i;
        if (t == 0) continue;                        // step t uses y_t, t>=1
        float Pm = Pf + QF;
        float K  = Pm / (Pm + Rf);
        float om = 1.0f - K;
        Ac = om * Ac;
        Bc = om * Bc + K * yv[i];
        Pf = om * Pm;
    }
    sA[tid] = Ac; sB[tid] = Bc;
    __syncthreads();

    // ---- block-wide inclusive affine scan (Hillis-Steele, 8 rounds) ----
    for (int off = 1; off < 256; off <<= 1) {
        float a2 = sA[tid], b2 = sB[tid];
        float a1 = 0.0f, b1 = 0.0f;
        const bool has = (tid >= off);
        if (has) { a1 = sA[tid - off]; b1 = sB[tid - off]; }
        __syncthreads();
        if (has) { sA[tid] = a2 * a1; sB[tid] = a2 * b1 + b2; }
        __syncthreads();
    }

    float x = (tid == 0) ? x0 : (sA[tid - 1] * x0 + sB[tid - 1]);

    // ---- pass 2: replay chunk, emit outputs (float4 stores) ----
    Pf = (float)P_start;
    float ov[4];
    #pragma unroll
    for (int i = 0; i < 32; ++i) {
        const int t = t0 + i;
        if (t != 0) {
            float Pm = Pf + QF;
            float K  = Pm / (Pm + Rf);
            x  = x + K * (yv[i] - x);
            Pf = (1.0f - K) * Pm;
        }
        ov[i & 3] = x;                               // out[0] = x0
        if ((i & 3) == 3) {
            float4 o; o.x = ov[0]; o.y = ov[1]; o.z = ov[2]; o.w = ov[3];
            *reinterpret_cast<float4*>(out + base + t0 + (i - 3)) = o;
        }
    }
}

extern "C" void kernel_launch(void* const* d_in, const int* in_sizes, int n_in,
                              void* d_out, int out_size, void* d_ws, size_t ws_size,
                              hipStream_t stream) {
    const float* in  = (const float*)d_in[0];
    float*       out = (float*)d_out;
    float*       Rws  = (float*)d_ws;            // [B_ROWS] Kalman R per row
    float*       x0ws = Rws + B_ROWS;            // [B_ROWS] median per row

    stats_median_kernel<<<B_ROWS, 256, 0, stream>>>(in, Rws, x0ws);
    kalman_scan_kernel<<<B_ROWS, 256, 0, stream>>>(in, Rws, x0ws, out);
}